// MoEGate_task_85718957294270
// MI455X (gfx1250) — compile-verified
//
#include <hip/hip_runtime.h>
#include <hip/hip_bf16.h>

typedef __attribute__((ext_vector_type(2))) float v2f;
typedef __attribute__((ext_vector_type(8))) float v8f;
typedef __attribute__((ext_vector_type(2))) float f32x2;
typedef __attribute__((ext_vector_type(4))) float f32x4;

// ---------------------------------------------------------------------------
// WMMA helper: one 16x16 f32 tile of D = A @ B^T using V_WMMA_F32_16X16X4_F32.
// A is row-major MxK (lda), BT holds W row-major NxK (ldb) so B[k][n] = BT[n*ldb+k].
// Uniform trip count (k4 loop) so no EXEC divergence around the WMMAs.
// Per-lane fragment layout (CDNA5 ISA 7.12.2):
//   A: m = lane%16, holds K = 4*k4 + 2*(lane/16) + {0,1}
//   B: n = lane%16, same k pairing
//   D: vgpr i -> row = i + 8*(lane/16), col = lane%16
// ---------------------------------------------------------------------------
__device__ inline v8f wmma_AxBT(const float* A, int lda, const float* BT, int ldb,
                                int mi, int ni, int K, v8f acc, int lane) {
  const int r = lane & 15;
  const int h = lane >> 4;
#pragma unroll
  for (int k4 = 0; k4 < K / 4; ++k4) {
    const int kb = 4 * k4 + 2 * h;
    v2f a; a.x = A[(mi + r) * lda + kb];  a.y = A[(mi + r) * lda + kb + 1];
    v2f b; b.x = BT[(ni + r) * ldb + kb]; b.y = BT[(ni + r) * ldb + kb + 1];
    acc = __builtin_amdgcn_wmma_f32_16x16x4_f32(false, a, false, b,
                                                (short)0, acc, false, false);
  }
  return acc;
}

// ---------------------------------------------------------------------------
// Precompute: clean[16][64] -> ws[0..1023], std[16][64] -> ws[1024..2047]
// (rows 0..5 valid). Single wave of 32 (EXEC all ones for WMMA).
// ---------------------------------------------------------------------------
__global__ __launch_bounds__(32) void moe_precompute(
    const float* __restrict__ embed, const float* __restrict__ ekeys,
    const float* __restrict__ Wq, const float* __restrict__ bq,
    const float* __restrict__ Wk, const float* __restrict__ bk,
    const float* __restrict__ Wg, const float* __restrict__ bg,
    const float* __restrict__ Wn, const float* __restrict__ bn,
    float* __restrict__ ws) {
  __shared__ float te[16 * 32];    // padded task embeddings
  __shared__ float kmat[32 * 32];  // K = expert_keys @ Wk^T + bk
  __shared__ float qmat[16 * 32];  // Q = (te @ Wq^T + bq) * scale
  __shared__ float ew[16 * 32];    // expert weights, rows >= 6 zero

  const int lane = threadIdx.x;
  const int r = lane & 15;
  const int h = lane >> 4;

  for (int m = 0; m < 16; ++m)
    te[m * 32 + lane] = (m < 6) ? embed[m * 32 + lane] : 0.0f;
  __syncthreads();

  // K matrix: 32x32, K-dim 32  -> 4 tiles x 8 wmma
  for (int mi = 0; mi < 32; mi += 16)
    for (int ni = 0; ni < 32; ni += 16) {
      v8f acc = {0, 0, 0, 0, 0, 0, 0, 0};
      acc = wmma_AxBT(ekeys, 32, Wk, 32, mi, ni, 32, acc, lane);
      for (int i = 0; i < 8; ++i)
        kmat[(mi + i + 8 * h) * 32 + ni + r] = acc[i] + bk[ni + r];
    }

  // Q: 16x32 (rows 0..5 meaningful), scaled by 1/sqrt(D=8)
  const float scale = 0.35355339059327373f;
  for (int ni = 0; ni < 32; ni += 16) {
    v8f acc = {0, 0, 0, 0, 0, 0, 0, 0};
    acc = wmma_AxBT(te, 32, Wq, 32, 0, ni, 32, acc, lane);
    for (int i = 0; i < 8; ++i)
      qmat[(i + 8 * h) * 32 + ni + r] = (acc[i] + bq[ni + r]) * scale;
  }
  __syncthreads();

  // 4-head attention over 32 expert keys + softmaxes: lanes 0..5, one task each
  for (int m = 6; m < 16; ++m) ew[m * 32 + lane] = 0.0f;
  if (lane < 6) {
    const int t = lane;
    float aw[32];
    for (int s = 0; s < 32; ++s) aw[s] = 0.0f;
    for (int hh = 0; hh < 4; ++hh) {
      float sc[32], mx = -1e30f;
      for (int s = 0; s < 32; ++s) {
        float d = 0.0f;
        for (int dd = 0; dd < 8; ++dd)
          d += qmat[t * 32 + hh * 8 + dd] * kmat[s * 32 + hh * 8 + dd];
        sc[s] = d; mx = fmaxf(mx, d);
      }
      float sum = 0.0f;
      for (int s = 0; s < 32; ++s) { sc[s] = __expf(sc[s] - mx); sum += sc[s]; }
      const float inv = 0.25f / sum;   // includes mean over 4 heads
      for (int s = 0; s < 32; ++s) aw[s] += sc[s] * inv;
    }
    float mx = -1e30f;
    for (int s = 0; s < 32; ++s) mx = fmaxf(mx, aw[s]);
    float sum = 0.0f;
    for (int s = 0; s < 32; ++s) { aw[s] = __expf(aw[s] - mx); sum += aw[s]; }
    const float inv = 1.0f / sum;
    for (int s = 0; s < 32; ++s) ew[t * 32 + s] = aw[s] * inv;
  }
  __syncthreads();

  // clean = ew @ Wg^T + bg  -> ws[0..1023]
  for (int ni = 0; ni < 64; ni += 16) {
    v8f acc = {0, 0, 0, 0, 0, 0, 0, 0};
    acc = wmma_AxBT(ew, 32, Wg, 32, 0, ni, 32, acc, lane);
    for (int i = 0; i < 8; ++i)
      ws[(i + 8 * h) * 64 + ni + r] = acc[i] + bg[ni + r];
  }
  // std = softplus(ew @ Wn^T + bn) + 0.01 -> ws[1024..2047]
  for (int ni = 0; ni < 64; ni += 16) {
    v8f acc = {0, 0, 0, 0, 0, 0, 0, 0};
    acc = wmma_AxBT(ew, 32, Wn, 32, 0, ni, 32, acc, lane);
    for (int i = 0; i < 8; ++i) {
      float x = acc[i] + bn[ni + r];
      float sp = (x > 20.0f) ? x : log1pf(__expf(x));
      ws[1024 + (i + 8 * h) * 64 + ni + r] = sp + 0.01f;
    }
  }
}

// ---------------------------------------------------------------------------
// Main streaming kernel: 128 threads (4 waves), 1 token/thread.
// Memory-bound: 128 MB noise in (NT), 128 MB gates out (NT).
// ---------------------------------------------------------------------------
#define TPB 128
__global__ __launch_bounds__(TPB) void moe_gate_main(
    const int* __restrict__ taskID, const float* __restrict__ noise,
    const float* __restrict__ ws, float* __restrict__ partials,
    float* __restrict__ out) {
  __shared__ __align__(16) float tbl[6 * 132];        // interleaved clean/std, stride 132 (conflict-free, 16B rows)
  __shared__ __align__(16) float nstage[4 * 32 * 66]; // per-wave staging, stride 66 -> bank (2l+j)%64
  __shared__ int   ri0[TPB]; __shared__ int ri1[TPB];
  __shared__ float rg0[TPB]; __shared__ float rg1[TPB];

  const int tid  = threadIdx.x;
  const int lane = tid & 31;
  const int wave = tid >> 5;

  // Load tables interleaved: tbl[t][2j] = clean, tbl[t][2j+1] = std
  for (int i = tid; i < 6 * 64; i += TPB) {
    const int t = i >> 6, j = i & 63;
    tbl[t * 132 + 2 * j]     = ws[i];
    tbl[t * 132 + 2 * j + 1] = ws[1024 + i];
  }
  __syncthreads();

  const int base = blockIdx.x * TPB + wave * 32;
  float* ns = &nstage[wave * 32 * 66];

  // Stage 32 noise rows: each iteration = one full 256B row, fully coalesced, NT
  for (int t = 0; t < 32; ++t) {
    f32x2 v = __builtin_nontemporal_load(
        (const f32x2*)(noise + (size_t)(base + t) * 64 + lane * 2));
    *(f32x2*)(ns + t * 66 + lane * 2) = v;
  }

  // Per-lane token: logits = clean + noise*std, top-2 scan
  const int task = taskID[base + lane];
  const float* tb  = &tbl[task * 132];
  const float* myn = ns + lane * 66;
  float v0 = -3.4e38f, v1 = -3.4e38f;
  int b0 = 0, b1 = 0;
  for (int j = 0; j < 64; j += 2) {
    f32x2 nn = *(const f32x2*)(myn + j);
    f32x4 cs = *(const f32x4*)(tb + 2 * j);   // (clean_j, std_j, clean_j1, std_j1)
    float l0 = fmaf(nn.x, cs.y, cs.x);
    float l1 = fmaf(nn.y, cs.w, cs.z);
    if (l0 > v0)      { v1 = v0; b1 = b0; v0 = l0; b0 = j; }
    else if (l0 > v1) { v1 = l0; b1 = j; }
    if (l1 > v0)      { v1 = v0; b1 = b0; v0 = l1; b0 = j + 1; }
    else if (l1 > v1) { v1 = l1; b1 = j + 1; }
  }
  const float g0 = 1.0f / (1.0f + __expf(v1 - v0));  // softmax over top-2
  const float g1 = 1.0f - g0;
  ri0[tid] = b0; ri1[tid] = b1; rg0[tid] = g0; rg1[tid] = g1;
  __syncthreads();

  // Store gates: 16 iters, 2 rows/iter, float4/lane, non-temporal streaming
  const int rrow = lane >> 4;
  const int j0   = (lane & 15) * 4;
  for (int it = 0; it < 16; ++it) {
    const int row = it * 2 + rrow;
    const int rr  = wave * 32 + row;
    const int a0 = ri0[rr], a1 = ri1[rr];
    const float p0 = rg0[rr], p1 = rg1[rr];
    f32x4 vv;
    vv.x = (j0     == a0) ? p0 : (j0     == a1) ? p1 : 0.0f;
    vv.y = (j0 + 1 == a0) ? p0 : (j0 + 1 == a1) ? p1 : 0.0f;
    vv.z = (j0 + 2 == a0) ? p0 : (j0 + 2 == a1) ? p1 : 0.0f;
    vv.w = (j0 + 3 == a0) ? p0 : (j0 + 3 == a1) ? p1 : 0.0f;
    __builtin_nontemporal_store(vv, (f32x4*)(out + (size_t)(base + row) * 64 + j0));
  }

  // Deterministic per-block load partial: fixed scan order, no FP atomics
  if (tid < 64) {
    float s = 0.0f;
    for (int i = 0; i < TPB; ++i) {
      if (ri0[i] == tid) s += rg0[i];
      if (ri1[i] == tid) s += rg1[i];
    }
    partials[(size_t)blockIdx.x * 64 + tid] = s;
  }
}

// Deterministic final reduction of per-block partials (fixed block order)
__global__ __launch_bounds__(64) void moe_reduce_load(
    const float* __restrict__ part, float* __restrict__ loadOut, int nblk) {
  const int j = threadIdx.x;
  float s = 0.0f;
  for (int b = 0; b < nblk; ++b) s += part[(size_t)b * 64 + j];
  loadOut[j] = s;
}

extern "C" void kernel_launch(void* const* d_in, const int* in_sizes, int n_in,
                              void* d_out, int out_size, void* d_ws, size_t ws_size,
                              hipStream_t stream) {
  const int*   taskID = (const int*)d_in[0];
  const float* noise  = (const float*)d_in[1];
  const float* embed  = (const float*)d_in[2];
  const float* ekeys  = (const float*)d_in[3];
  const float* Wq     = (const float*)d_in[4];
  const float* bq     = (const float*)d_in[5];
  const float* Wk     = (const float*)d_in[6];
  const float* bk     = (const float*)d_in[7];
  const float* Wg     = (const float*)d_in[8];
  const float* bg     = (const float*)d_in[9];
  const float* Wn     = (const float*)d_in[10];
  const float* bn     = (const float*)d_in[11];
  float* out = (float*)d_out;
  float* ws  = (float*)d_ws;

  const int Bn    = in_sizes[0];      // 524288
  const int nblk  = Bn / TPB;         // 4096
  float* partials = ws + 2048;        // after clean/std tables

  moe_precompute<<<1, 32, 0, stream>>>(embed, ekeys, Wq, bq, Wk, bk,
                                       Wg, bg, Wn, bn, ws);
  moe_gate_main<<<nblk, TPB, 0, stream>>>(taskID, noise, ws, partials, out);
  moe_reduce_load<<<1, 64, 0, stream>>>(partials, out + (size_t)Bn * 64, nblk);
}